// GraphBlock_1949915152398
// MI455X (gfx1250) — compile-verified
//
#include <hip/hip_runtime.h>

typedef __attribute__((ext_vector_type(2))) float v2f;
typedef __attribute__((ext_vector_type(8))) float v8f;

#define HID 128
#define WT_STRIDE 132   // 132 mod 64 = 4 -> conflict-free b64 LDS reads across 32 lanes
#define BN_EPS 1e-5f

// Native f32 global atomic add (no-return form -> STOREcnt). Avoids CAS loop.
__device__ __forceinline__ void atomic_add_f32_dev(float* addr, float val) {
  asm volatile("global_atomic_add_f32 %0, %1, off"
               :: "v"(addr), "v"(val) : "memory");
}

// ---------------------------------------------------------------------------
// K1: zero degree counters and BN stat bins
__global__ void init_misc(unsigned* __restrict__ deg, float* __restrict__ stats, int N) {
  long i = (long)blockIdx.x * blockDim.x + threadIdx.x;
  if (i < N) deg[i] = 0u;
  if (i < 256) stats[i] = 0.0f;   // sums[128] + sumsq[128]
}

// K2: degree = count of dst occurrences (self-loop added later as +1)
__global__ void count_deg(const long long* __restrict__ dst, unsigned* __restrict__ deg, int E) {
  long i = (long)blockIdx.x * blockDim.x + threadIdx.x;
  if (i < E) atomicAdd(&deg[(long)dst[i]], 1u);
}

// K3: dinv = rsqrt(deg + 1)  (deg >= 1 always due to self loop)
__global__ void calc_dinv(const unsigned* __restrict__ deg, float* __restrict__ dinv, int N) {
  long i = (long)blockIdx.x * blockDim.x + threadIdx.x;
  if (i < N) dinv[i] = rsqrtf((float)(deg[i] + 1u));
}

// ---------------------------------------------------------------------------
// K4: h = x @ W via V_WMMA_F32_16X16X4_F32.
// Block = 256 threads = 8 waves; each wave computes a 16-row x 128-col stripe
// (8 accumulators of 16x16). W staged TRANSPOSED in LDS with padded stride so
// each lane's (K, K+1) pair is one aligned ds_load_b64 and banks don't collide.
__global__ void __launch_bounds__(256) gemm_wmma_f32(
    const float* __restrict__ x, const float* __restrict__ W,
    float* __restrict__ h, int N) {
  __shared__ float Wt[HID * WT_STRIDE];   // ~66 KB of the 320 KB WGP LDS
  const int tid = threadIdx.x;
#pragma unroll
  for (int it = 0; it < (HID * HID) / 256; ++it) {
    int idx = tid + it * 256;             // coalesced global read of W[k][j]
    int kk = idx >> 7;
    int j  = idx & 127;
    Wt[j * WT_STRIDE + kk] = W[idx];      // store transposed: Wt[j][k]
  }
  __syncthreads();

  const int wave = tid >> 5;
  const int lane = tid & 31;
  const int m    = lane & 15;           // row/col within 16-tile
  const int kh   = (lane >> 4) << 1;    // K sub-offset: 0 or 2

  const long rowbase = (long)blockIdx.x * 128 + (long)wave * 16;
  long arow = rowbase + m;
  if (arow >= N) arow = N - 1;          // clamp: keep EXEC all-ones for WMMA
  const float* xr = x + arow * HID;

  v8f acc[8] = {};

  for (int k = 0; k < HID; k += 4) {
    // A tile 16x4: lane(0-15)=M, VGPR0/1 = K {kh, kh+1}; 8B aligned -> b64 load
    v2f a = *(const v2f*)(xr + k + kh);
#pragma unroll
    for (int t = 0; t < 8; ++t) {
      // B tile 4x16: lane(0-15)=N cols; Wt[col][k+kh], Wt[col][k+kh+1] contiguous
      v2f bb = *(const v2f*)(&Wt[(t * 16 + m) * WT_STRIDE + k + kh]);
      acc[t] = __builtin_amdgcn_wmma_f32_16x16x4_f32(
          false, a, false, bb, (short)0, acc[t], false, false);
    }
  }

  // D layout: VGPR r -> M = r + (lane<16 ? 0 : 8), N = lane&15
  const int mr = (lane >> 4) * 8;
#pragma unroll
  for (int t = 0; t < 8; ++t) {
#pragma unroll
    for (int r = 0; r < 8; ++r) {
      long orow = rowbase + mr + r;
      if (orow < N) h[orow * HID + t * 16 + m] = acc[t][r];
    }
  }
}

// ---------------------------------------------------------------------------
// K5: agg = bias + self-loop contribution h[i] * dinv[i]^2
__global__ void agg_init(const float* __restrict__ hh, const float* __restrict__ dinv,
                         const float* __restrict__ bias, float* __restrict__ agg,
                         long total) {
  long i = (long)blockIdx.x * blockDim.x + threadIdx.x;
  if (i >= total) return;
  long r = i >> 7;
  int  c = (int)(i & 127);
  float di = dinv[r];
  agg[i] = fmaf(hh[i], di * di, bias[c]);
}

// K6: edge scatter. One wave per edge; float4 per lane (512B coalesced row read),
// 4 native f32 atomics per lane into agg[dst].
__global__ void __launch_bounds__(256) scatter_edges(
    const long long* __restrict__ src, const long long* __restrict__ dst,
    const float* __restrict__ dinv, const float* __restrict__ hh,
    float* __restrict__ agg, int E) {
  int e = blockIdx.x * 8 + (threadIdx.x >> 5);
  if (e >= E) return;
  const int lane = threadIdx.x & 31;
  const long s = (long)src[e];
  const long d = (long)dst[e];
  const float nrm = dinv[s] * dinv[d];
  const float4 v = ((const float4*)(hh + s * HID))[lane];
  float* base = agg + d * HID + (long)lane * 4;
  atomic_add_f32_dev(base + 0, v.x * nrm);
  atomic_add_f32_dev(base + 1, v.y * nrm);
  atomic_add_f32_dev(base + 2, v.z * nrm);
  atomic_add_f32_dev(base + 3, v.w * nrm);
}

// ---------------------------------------------------------------------------
// K7: per-column partial sums/sumsq; rows partitioned by blockIdx (each row read
// once). Prefetch the next-stride row (global_prefetch_b8) to hide latency.
__global__ void __launch_bounds__(128) bn_partial(
    const float* __restrict__ agg, float* __restrict__ sums,
    float* __restrict__ sumsq, int N) {
  const int c = threadIdx.x;   // 128 columns
  float s = 0.0f, q = 0.0f;
  for (int r = blockIdx.x; r < N; r += gridDim.x) {
    long nr = (long)r + gridDim.x;
    if (nr < N) __builtin_prefetch(&agg[nr * HID + c], 0, 1);
    float v = agg[(long)r * HID + c];
    s += v;
    q = fmaf(v, v, q);
  }
  atomic_add_f32_dev(&sums[c], s);
  atomic_add_f32_dev(&sumsq[c], q);
}

// K8: fold mean/var/gamma/beta into per-column scale & shift
__global__ void bn_coeffs(const float* __restrict__ sums, const float* __restrict__ sumsq,
                          const float* __restrict__ gamma, const float* __restrict__ beta,
                          float* __restrict__ scale, float* __restrict__ shift, float invN) {
  int c = threadIdx.x;
  float mean = sums[c] * invN;
  float var  = fmaf(-mean, mean, sumsq[c] * invN);  // biased variance
  float sc   = gamma[c] * rsqrtf(var + BN_EPS);
  scale[c] = sc;
  shift[c] = fmaf(-mean, sc, beta[c]);
}

// K9: out = relu(agg*scale + shift) + last_x
__global__ void epilogue(const float* __restrict__ agg, const float* __restrict__ scale,
                         const float* __restrict__ shift, const float* __restrict__ lx,
                         float* __restrict__ out, long total) {
  long i = (long)blockIdx.x * blockDim.x + threadIdx.x;
  if (i >= total) return;
  int c = (int)(i & 127);
  float v = fmaf(agg[i], scale[c], shift[c]);
  out[i] = fmaxf(v, 0.0f) + lx[i];
}

// ---------------------------------------------------------------------------
extern "C" void kernel_launch(void* const* d_in, const int* in_sizes, int n_in,
                              void* d_out, int out_size, void* d_ws, size_t ws_size,
                              hipStream_t stream) {
  (void)n_in; (void)out_size; (void)ws_size;

  const float*     x      = (const float*)d_in[0];
  const float*     last_x = (const float*)d_in[1];
  const long long* ei     = (const long long*)d_in[2];   // [2, E] int64
  const float*     W      = (const float*)d_in[3];
  const float*     bias   = (const float*)d_in[4];
  const float*     gamma  = (const float*)d_in[5];
  const float*     beta   = (const float*)d_in[6];

  const int  N     = in_sizes[0] / HID;   // 50000
  const int  E     = in_sizes[2] / 2;     // 800000
  const long total = (long)N * HID;

  // workspace layout (floats): h[N*128] | agg[N*128] | dinv[N] | deg[N] | stats[512]
  float*    h    = (float*)d_ws;
  float*    agg  = h + total;
  float*    dinv = agg + total;
  unsigned* deg  = (unsigned*)(dinv + N);
  float*    stats = (float*)(deg + N);
  float*    sums  = stats;
  float*    sumsq = stats + HID;
  float*    scale = stats + 2 * HID;
  float*    shift = stats + 3 * HID;

  const int TPB = 256;
  init_misc   <<<(N + TPB - 1) / TPB, TPB, 0, stream>>>(deg, stats, N);
  count_deg   <<<(E + TPB - 1) / TPB, TPB, 0, stream>>>(ei + E, deg, E);
  calc_dinv   <<<(N + TPB - 1) / TPB, TPB, 0, stream>>>(deg, dinv, N);
  gemm_wmma_f32<<<(N + 127) / 128, 256, 0, stream>>>(x, W, h, N);
  agg_init    <<<(int)((total + TPB - 1) / TPB), TPB, 0, stream>>>(h, dinv, bias, agg, total);
  scatter_edges<<<(E + 7) / 8, 256, 0, stream>>>(ei, ei + E, dinv, h, agg, E);
  bn_partial  <<<256, 128, 0, stream>>>(agg, sums, sumsq, N);
  bn_coeffs   <<<1, 128, 0, stream>>>(sums, sumsq, gamma, beta, scale, shift, 1.0f / (float)N);
  epilogue    <<<(int)((total + TPB - 1) / TPB), TPB, 0, stream>>>(agg, scale, shift, last_x,
                                                                   (float*)d_out, total);
}